// Skipgram_17987323036328
// MI455X (gfx1250) — compile-verified
//
#include <hip/hip_runtime.h>

// Skip-gram negative-sampling loss for MI455X (gfx1250, wave32).
//
// Memory-bound problem (~100 MB gathered reads, ~50 MFLOP): stream gathered
// embedding rows once each; do the batched dot products with chained
// V_WMMA_F32_16X16X4_F32 (16 batch elements per wave, diagonal of the 16x16
// product = the 16 per-element dot products). Two independent WMMA
// accumulator chains (pos-score, summed-neg-score) per wave.

typedef __attribute__((ext_vector_type(2))) float v2f;
typedef __attribute__((ext_vector_type(8))) float v8f;

#define DIM   128
#define KNEG  10
#define TILE  16           // batch elements per wave (WMMA M=N=16)
#define WPB   8            // waves per block
#define BLOCK_THREADS (WPB * 32)

__device__ __forceinline__ float log_sigmoid_f(float x) {
    // stable: log(sigmoid(x)) = min(x,0) - log1p(exp(-|x|))
    return fminf(x, 0.0f) - log1pf(expf(-fabsf(x)));
}

__global__ __launch_bounds__(BLOCK_THREADS)
void skipgram_wmma_kernel(const int* __restrict__ center_id,
                          const int* __restrict__ pos_id,
                          const int* __restrict__ neg_ids,
                          const float* __restrict__ W_in,
                          const float* __restrict__ W_out,
                          float* __restrict__ block_partials)
{
    const int lane = threadIdx.x & 31;
    const int wave = threadIdx.x >> 5;
    const int tile = blockIdx.x * WPB + wave;

    // WMMA 16x16x4 f32 operand layout (wave32):
    //   A (16x4 MxK):  lane L holds A[L&15][2*(L>>4) + {0,1}] in v2f
    //   B (4x16 KxN):  lane L holds B[2*(L>>4) + {0,1}][L&15] in v2f
    // Both reduce to: element [row = L&15][dim = 4k + 2*(L>>4) + {0,1}]
    const int m   = lane & 15;          // batch row within tile (and N column)
    const int off = (lane >> 4) << 1;   // 0 or 2: K sub-offset for this half-wave
    const int b   = tile * TILE + m;

    const float* crow = W_in  + (size_t)center_id[b] * DIM;
    const float* prow = W_out + (size_t)pos_id[b]    * DIM;
    const float* nrow[KNEG];
#pragma unroll
    for (int j = 0; j < KNEG; ++j)
        nrow[j] = W_out + (size_t)neg_ids[b * KNEG + j] * DIM;

    v8f cpos = {};   // accumulates center . pos     (diag holds the dots)
    v8f cneg = {};   // accumulates center . sum_neg

#pragma unroll 4
    for (int k = 0; k < DIM / 4; ++k) {
        const int d = 4 * k + off;
        v2f a  = *(const v2f*)(crow + d);
        v2f bp = *(const v2f*)(prow + d);
        v2f bn; bn.x = 0.0f; bn.y = 0.0f;
#pragma unroll
        for (int j = 0; j < KNEG; ++j) {
            v2f t = *(const v2f*)(nrow[j] + d);
            bn.x += t.x; bn.y += t.y;
        }
        // 8-arg form: (neg_a, A, neg_b, B, c_mod, C, reuse_a, reuse_b)
        cpos = __builtin_amdgcn_wmma_f32_16x16x4_f32(
                   false, a, false, bp, (short)0, cpos, false, false);
        cneg = __builtin_amdgcn_wmma_f32_16x16x4_f32(
                   false, a, false, bn, (short)0, cneg, false, false);
    }

    // Diagonal extraction from 16x16 f32 C/D layout:
    //   diag[i], i<8 : VGPR i,   lane i          (lanes 0..7)
    //   diag[8+r]    : VGPR r,   lane 24+r       (lanes 24..31)
    float sp = 0.0f, sn = 0.0f;
#pragma unroll
    for (int r = 0; r < 8; ++r) {
        if (lane == r)      { sp = cpos[r]; sn = cneg[r]; }
        if (lane == 24 + r) { sp = cpos[r]; sn = cneg[r]; }
    }

    float contrib = 0.0f;
    if (lane < 8 || lane >= 24)
        contrib = log_sigmoid_f(sp) + log_sigmoid_f(-sn);

    // wave32 butterfly reduction
#pragma unroll
    for (int s = 16; s > 0; s >>= 1)
        contrib += __shfl_xor(contrib, s, 32);

    __shared__ float wsum[WPB];
    if (lane == 0) wsum[wave] = contrib;
    __syncthreads();

    if (threadIdx.x == 0) {
        float t = 0.0f;
#pragma unroll
        for (int w = 0; w < WPB; ++w) t += wsum[w];
        block_partials[blockIdx.x] = t;
    }
}

// Deterministic fixed-order final reduction; writes the negated loss.
__global__ __launch_bounds__(256)
void skipgram_reduce_kernel(const float* __restrict__ partials, int n,
                            float* __restrict__ out)
{
    __shared__ float sm[256];
    float t = 0.0f;
    for (int i = threadIdx.x; i < n; i += 256) t += partials[i];
    sm[threadIdx.x] = t;
    __syncthreads();
    for (int s = 128; s > 0; s >>= 1) {
        if (threadIdx.x < s) sm[threadIdx.x] += sm[threadIdx.x + s];
        __syncthreads();
    }
    if (threadIdx.x == 0) out[0] = -sm[0];
}

extern "C" void kernel_launch(void* const* d_in, const int* in_sizes, int n_in,
                              void* d_out, int out_size, void* d_ws, size_t ws_size,
                              hipStream_t stream)
{
    const int*   center_id = (const int*)  d_in[0];
    const int*   pos_id    = (const int*)  d_in[1];
    const int*   neg_ids   = (const int*)  d_in[2];
    const float* W_in      = (const float*)d_in[3];
    const float* W_out     = (const float*)d_in[4];
    float*       out       = (float*)d_out;
    float*       partials  = (float*)d_ws;

    const int B      = in_sizes[0];                 // 16384
    const int tiles  = B / TILE;                    // 1024 waves
    const int blocks = tiles / WPB;                 // 128 blocks

    skipgram_wmma_kernel<<<blocks, BLOCK_THREADS, 0, stream>>>(
        center_id, pos_id, neg_ids, W_in, W_out, partials);
    skipgram_reduce_kernel<<<1, 256, 0, stream>>>(partials, blocks, out);
}